// ContextAwareFlowRouter_22101901705432
// MI455X (gfx1250) — compile-verified
//
#include <hip/hip_runtime.h>
#include <hip/hip_bf16.h>

// ---------------- problem constants ----------------
constexpr int NB   = 4;
constexpr int NS   = 1024;
constexpr int TN   = NB * NS;     // 4096 tokens
constexpr int NIN  = 128;
constexpr int NOUT = 128;
constexpr int NP   = 16;
constexpr int NCOMB= 256;
constexpr int ND   = NOUT * NIN;  // 16384 elements per token
constexpr float MAXSEQ = 4096.f;

typedef float v2f __attribute__((ext_vector_type(2)));
typedef float v8f __attribute__((ext_vector_type(8)));

__device__ __forceinline__ float gelu_erf(float v){
    return 0.5f * v * (1.f + erff(v * 0.70710678118654752f));
}
__device__ __forceinline__ float sigm(float v){
    return 1.f / (1.f + expf(-v));
}

// ---------------------------------------------------------------------------
// K1: per-token MLPs -> pwi[t][p] = softmax(sel)(p) * sigmoid(intensity)
// one block (64 threads) per token; weights stream from L2 (hot, ~100 KB total)
// ---------------------------------------------------------------------------
__global__ void k1_token_mlps(const float* __restrict__ x, const float* __restrict__ ctx,
    const float* __restrict__ w_sel1, const float* __restrict__ b_sel1,
    const float* __restrict__ w_sel2, const float* __restrict__ b_sel2,
    const float* __restrict__ w_int1, const float* __restrict__ b_int1,
    const float* __restrict__ w_int2, const float* __restrict__ b_int2,
    float* __restrict__ pwi)
{
    __shared__ float comb[NCOMB];
    __shared__ float h1g[32];
    __shared__ float sm[16];
    __shared__ float red[64];
    __shared__ float inten_s;

    const int t   = blockIdx.x;
    const int tid = threadIdx.x;           // 64 threads

    for (int i = tid; i < 128; i += 64){
        comb[i]       = x[t * 128 + i];
        comb[128 + i] = ctx[t * 128 + i];
    }
    __syncthreads();

    // intensity hidden layer (64 units), all threads
    {
        float acc = b_int1[tid];
        for (int k = 0; k < NCOMB; ++k) acc = fmaf(comb[k], w_int1[k * 64 + tid], acc);
        red[tid] = gelu_erf(acc) * w_int2[tid];
    }
    // selector hidden layer (32 units)
    if (tid < 32){
        float acc = b_sel1[tid];
        for (int k = 0; k < NCOMB; ++k) acc = fmaf(comb[k], w_sel1[k * 32 + tid], acc);
        h1g[tid] = gelu_erf(acc);
    }
    __syncthreads();

    // selector logits (16)
    if (tid < 16){
        float acc = b_sel2[tid];
        for (int k = 0; k < 32; ++k) acc = fmaf(h1g[k], w_sel2[k * 16 + tid], acc);
        sm[tid] = acc;
    }
    // reduce intensity dot product
    for (int off = 32; off > 0; off >>= 1){
        if (tid < off) red[tid] += red[tid + off];
        __syncthreads();
    }
    if (tid == 0) inten_s = sigm(red[0] + b_int2[0]);
    __syncthreads();

    if (tid < 16){
        float m = sm[0];
        for (int j = 1; j < 16; ++j) m = fmaxf(m, sm[j]);
        float ssum = 0.f;
        for (int j = 0; j < 16; ++j) ssum += expf(sm[j] - m);
        float p = expf(sm[tid] - m) / ssum;
        pwi[t * NP + tid] = p * inten_s;   // fold intensity into the GEMM A-matrix
    }
}

// ---------------------------------------------------------------------------
// K2a: partial sums of x over the sequence dim -> partial[b][g][j]
// ---------------------------------------------------------------------------
__global__ void k2a_xmean_partial(const float* __restrict__ x, float* __restrict__ partial)
{
    __shared__ float tmp[2][128];
    const int b = blockIdx.x, g = blockIdx.y;
    const int tid = threadIdx.x;           // 256
    const int j = tid & 127, hv = tid >> 7;
    float s = 0.f;
    const int s0 = g * 128 + hv * 64;
    for (int si = 0; si < 64; ++si) s += x[((b << 10) + s0 + si) * 128 + j];
    tmp[hv][j] = s;
    __syncthreads();
    if (hv == 0) partial[(b * 8 + g) * 128 + j] = tmp[0][j] + tmp[1][j];
}

// ---------------------------------------------------------------------------
// K2b: window-size MLP per batch -> global scalar k (top-k count)
// ---------------------------------------------------------------------------
__global__ void k2b_kcalc(const float* __restrict__ partial,
    const float* __restrict__ w_win1, const float* __restrict__ b_win1,
    const float* __restrict__ w_win2, const float* __restrict__ b_win2,
    int* __restrict__ kval)
{
    __shared__ float mean[4][128];
    __shared__ float red[64];
    __shared__ float wsum_s;
    const int tid = threadIdx.x;           // 128
    for (int b = 0; b < 4; ++b){
        float s = 0.f;
        for (int g = 0; g < 8; ++g) s += partial[(b * 8 + g) * 128 + tid];
        mean[b][tid] = s * (1.f / 1024.f);
    }
    if (tid == 0) wsum_s = 0.f;
    __syncthreads();
    for (int b = 0; b < 4; ++b){
        if (tid < 64){
            float acc = b_win1[tid];
            for (int i = 0; i < 128; ++i) acc = fmaf(mean[b][i], w_win1[i * 64 + tid], acc);
            red[tid] = gelu_erf(acc) * w_win2[tid];
        }
        __syncthreads();
        for (int off = 32; off > 0; off >>= 1){
            if (tid < off) red[tid] += red[tid + off];
            __syncthreads();
        }
        if (tid == 0){
            float wsz = sigm(red[0] + b_win2[0]) * (MAXSEQ - 256.f) + 256.f;
            wsum_s += wsz;
        }
        __syncthreads();
    }
    if (tid == 0){
        float sp = 0.1f * (wsum_s * 0.25f) / MAXSEQ;
        float kf = floorf((float)ND * sp);
        int k = (int)kf;
        if (k < 1) k = 1;
        if (k > ND) k = ND;
        *kval = k;
    }
}

// ---------------------------------------------------------------------------
// K3: flow GEMM via V_WMMA_F32_16X16X4_F32.
//   D[16 tok x 16 e] = A[16 tok x 16 p] * B[16 p x 16 e], K=16 as 4 k-steps.
// grid (TN/16, 8), 256 threads = 8 waves; wave handles 16 tiles (256 e).
// Writes unmasked (intensity-folded) flow to d_out: 256 MB stream.
// ---------------------------------------------------------------------------
__global__ void __launch_bounds__(256)
k3_flow_wmma(const float* __restrict__ pwi, const float* __restrict__ fp,
             float* __restrict__ out)
{
    const int tid  = threadIdx.x;
    const int lane = tid & 31;
    const int wv   = tid >> 5;               // wave 0..7
    const int n    = lane & 15;              // tile column / A row
    const int hv   = lane >> 4;              // lane half
    const int t0   = blockIdx.x * 16;        // first token of tile
    const int eb0  = blockIdx.y * 2048 + wv * 256;

    // A tile (pattern_weights * intensity), constant across the wave's 16 tiles.
    // ISA layout A 16x4: lanes0-15 VGPR0=K0,VGPR1=K1; lanes16-31 VGPR0=K2,VGPR1=K3.
    v2f aT[4];
#pragma unroll
    for (int kk = 0; kk < 4; ++kk){
        const int k0 = kk * 4 + 2 * hv;
        aT[kk].x = pwi[(t0 + n) * NP + k0];
        aT[kk].y = pwi[(t0 + n) * NP + k0 + 1];
    }

#pragma unroll 1
    for (int tile = 0; tile < 16; ++tile){
        const int eb = eb0 + tile * 16;
        v8f c = {0.f, 0.f, 0.f, 0.f, 0.f, 0.f, 0.f, 0.f};
#pragma unroll
        for (int kk = 0; kk < 4; ++kk){
            const int k0 = kk * 4 + 2 * hv;
            v2f bT;
            bT.x = fp[(size_t)(k0    ) * ND + eb + n];
            bT.y = fp[(size_t)(k0 + 1) * ND + eb + n];
            c = __builtin_amdgcn_wmma_f32_16x16x4_f32(
                    false, aT[kk], false, bT, (short)0, c, false, false);
        }
        // C/D layout: lane holds column n, VGPR j = row (j + 8*hv)
#pragma unroll
        for (int j = 0; j < 8; ++j){
            out[(size_t)(t0 + j + 8 * hv) * ND + eb + n] = c[j];
        }
    }
}

// ---------------------------------------------------------------------------
// K4: per-token exact k-th-largest-|v| threshold via 4-pass 8-bit radix
// select on abs bit patterns (monotone for non-negative floats), all in LDS,
// then zero out elements below threshold. One 256-thread block per token.
// Dynamic LDS: row[16384] + 8 per-wave hists + totals + ctrl (~73 KB, OK on
// CDNA5's 320 KB WGP LDS).
// ---------------------------------------------------------------------------
__global__ void __launch_bounds__(256)
k4_select_mask(float* __restrict__ out, const int* __restrict__ kval)
{
    extern __shared__ unsigned smem[];
    unsigned* row   = smem;                  // 16384
    unsigned* whist = smem + ND;             // 8 * 256
    unsigned* tot   = whist + 8 * 256;       // 256
    unsigned* ctrl  = tot + 256;             // [0]=prefix bits, [1]=remaining rank

    const int tid = threadIdx.x;
    const size_t base = (size_t)blockIdx.x * ND;

    for (int i = tid; i < ND; i += 256) row[i] = __float_as_uint(out[base + i]);
    if (tid == 0){ ctrl[0] = 0u; ctrl[1] = (unsigned)(*kval); }
    __syncthreads();

    unsigned* mh = whist + (tid >> 5) * 256; // per-wave histogram copy

#pragma unroll 1
    for (int pass = 0; pass < 4; ++pass){
        for (int i = tid; i < 8 * 256; i += 256) whist[i] = 0u;
        __syncthreads();
        const int shift = 24 - 8 * pass;
        const unsigned pref   = ctrl[0];
        const unsigned himask = (pass == 0) ? 0u : (0xFFFFFFFFu << (shift + 8));
        for (int i = tid; i < ND; i += 256){
            const unsigned a = row[i] & 0x7FFFFFFFu;
            if ((a & himask) == pref) atomicAdd(&mh[(a >> shift) & 255u], 1u);
        }
        __syncthreads();
        {   // reduce the 8 wave copies
            unsigned s = 0;
#pragma unroll
            for (int w = 0; w < 8; ++w) s += whist[w * 256 + tid];
            tot[tid] = s;
        }
        __syncthreads();
        if (tid == 0){
            unsigned r = ctrl[1], acc = 0;
            for (int d = 255; d >= 0; --d){
                acc += tot[d];
                if (acc >= r){
                    ctrl[1] = r - (acc - tot[d]);
                    ctrl[0] = pref | ((unsigned)d << shift);
                    break;
                }
            }
        }
        __syncthreads();
    }

    const unsigned thr = ctrl[0];            // exact bits of k-th largest |v|
    for (int i = tid; i < ND; i += 256){
        const unsigned a = row[i] & 0x7FFFFFFFu;
        if (a < thr) out[base + i] = 0.f;    // kept values already in d_out
    }
}

// ---------------------------------------------------------------------------
extern "C" void kernel_launch(void* const* d_in, const int* in_sizes, int n_in,
                              void* d_out, int out_size, void* d_ws, size_t ws_size,
                              hipStream_t stream)
{
    (void)in_sizes; (void)n_in; (void)out_size; (void)ws_size;

    const float* x      = (const float*)d_in[0];
    const float* ctx    = (const float*)d_in[1];
    const float* fp     = (const float*)d_in[2];
    const float* w_sel1 = (const float*)d_in[3];
    const float* b_sel1 = (const float*)d_in[4];
    const float* w_sel2 = (const float*)d_in[5];
    const float* b_sel2 = (const float*)d_in[6];
    const float* w_int1 = (const float*)d_in[7];
    const float* b_int1 = (const float*)d_in[8];
    const float* w_int2 = (const float*)d_in[9];
    const float* b_int2 = (const float*)d_in[10];
    const float* w_win1 = (const float*)d_in[11];
    const float* b_win1 = (const float*)d_in[12];
    const float* w_win2 = (const float*)d_in[13];
    const float* b_win2 = (const float*)d_in[14];

    float* out     = (float*)d_out;
    float* pwi     = (float*)d_ws;               // 4096*16 floats
    float* partial = pwi + TN * NP;              // 4*8*128 floats
    int*   kval    = (int*)(partial + 4 * 8 * 128);

    k1_token_mlps<<<TN, 64, 0, stream>>>(x, ctx, w_sel1, b_sel1, w_sel2, b_sel2,
                                         w_int1, b_int1, w_int2, b_int2, pwi);
    k2a_xmean_partial<<<dim3(4, 8), 256, 0, stream>>>(x, partial);
    k2b_kcalc<<<1, 128, 0, stream>>>(partial, w_win1, b_win1, w_win2, b_win2, kval);
    k3_flow_wmma<<<dim3(TN / 16, 8), 256, 0, stream>>>(pwi, fp, out);

    const size_t lds_bytes = (size_t)(ND + 8 * 256 + 256 + 4) * sizeof(unsigned);
    k4_select_mask<<<TN, 256, lds_bytes, stream>>>(out, kval);
}